// convert_to_frequency_63977832841442
// MI455X (gfx1250) — compile-verified
//
#include <hip/hip_runtime.h>
#include <hip/hip_bf16.h>

// ---------------------------------------------------------------------------
// Re(FFT_512(x)) = X @ C,  C[n][k] = cos(2*pi*n*k/512).
// Symmetry C[n][k] = C[512-n][k] and out[k] = out[512-k] reduces the GEMM to
//   OUT[:, 0:256] = Y(32768 x 260) @ B(260 x 256)
// y[n] = x[n]+x[512-n] (n=1..255), y[0]=x[0], y[256]=x[256] (B-row 256 =
// (-1)^c rank-1 term), rows 257..259 zero-pad K to a multiple of 4.
// Cols 257..511 are mirror-stores of 255..1; col 256 = alternating sum.
// fp32 WMMA: V_WMMA_F32_16X16X4_F32.  2x2 register tiling per wave:
// each WMMA B-fragment is reused by two M-tiles (2 wmma per global b64).
// ---------------------------------------------------------------------------

typedef float v2f __attribute__((ext_vector_type(2)));
typedef float v4f __attribute__((ext_vector_type(4)));
typedef float v8f __attribute__((ext_vector_type(8)));
typedef int  vsi4 __attribute__((vector_size(16)));    // int4 for async builtin

#define NFFT       512
#define ROWS_TOTAL 32768
#define BLK_M      32            // rows per block (two 16-row M tiles)
#define KFOLD      260           // 256 folded rows + rank-1 row + 3 zero pad

#define __AS1__ __attribute__((address_space(1)))
#define __AS3__ __attribute__((address_space(3)))

// ---------------------------------------------------------------------------
// Bp fragment-native layout: idx = ((k0/4)*2 + h)*512 + c*2 + j
//   value = B[k0+2h+j][c];  B[r][c] = cos(2*pi*r*c/512) for r<=256, 0 for pads
// Total: 65*2*512 = 66560 floats = 266 KB in d_ws.
// ---------------------------------------------------------------------------
__global__ void build_dft_b_63977832841442(float* __restrict__ Bp) {
    int idx = blockIdx.x * blockDim.x + threadIdx.x;   // 0 .. 66559
    int j  = idx & 1;
    int c  = (idx >> 1) & 255;
    int h  = (idx >> 9) & 1;
    int r4 = idx >> 10;                                // 0..64
    int r  = r4 * 4 + 2 * h + j;                       // 0..259
    float v = 0.0f;
    if (r <= 256) {
        int t = (r * c) & 511;                         // exact in int
        v = cospif((float)t * (1.0f / 256.0f));        // cos(2*pi*t/512)
    }
    Bp[idx] = v;
}

// ---------------------------------------------------------------------------
// One block = 32-row M tile pair. 8 waves; wave w owns output cols
// [32w, 32w+32) as two 16-wide N tiles x both M tiles (4 accumulators).
// ---------------------------------------------------------------------------
__global__ __launch_bounds__(256) void dft_wmma_63977832841442(
        const float* __restrict__ A,
        const float* __restrict__ Bp,
        float* __restrict__ out) {
    __shared__ float sX[BLK_M * NFFT];     // 64 KB raw tile
    __shared__ float sY[BLK_M * KFOLD];    // 32.5 KB folded tile

    const int tid      = threadIdx.x;
    const int row_base = blockIdx.x * BLK_M;

    // ---- Stage raw 32x512 tile (contiguous 64 KB) into LDS ----------------
    {
        const float* g = A + (size_t)row_base * NFFT + tid * 4;
        float*       s = sX + tid * 4;
#if __has_builtin(__builtin_amdgcn_global_load_async_to_lds_b128)
        // CDNA5 async copy: ASYNCcnt-tracked, no VGPR round trip.
        __AS1__ vsi4* gp = (__AS1__ vsi4*)g;
        __AS3__ vsi4* sp = (__AS3__ vsi4*)s;
#define ALD_63977832841442(off) \
        __builtin_amdgcn_global_load_async_to_lds_b128(gp, sp, off, 0)
        ALD_63977832841442(0);     ALD_63977832841442(4096);
        ALD_63977832841442(8192);  ALD_63977832841442(12288);
        ALD_63977832841442(16384); ALD_63977832841442(20480);
        ALD_63977832841442(24576); ALD_63977832841442(28672);
        ALD_63977832841442(32768); ALD_63977832841442(36864);
        ALD_63977832841442(40960); ALD_63977832841442(45056);
        ALD_63977832841442(49152); ALD_63977832841442(53248);
        ALD_63977832841442(57344); ALD_63977832841442(61440);
#undef ALD_63977832841442
#if __has_builtin(__builtin_amdgcn_s_wait_asynccnt)
        __builtin_amdgcn_s_wait_asynccnt(0);
#else
        asm volatile("s_wait_asynccnt 0" ::: "memory");
#endif
#else
#pragma unroll
        for (int i = 0; i < 16; ++i)
            *(v4f*)(s + i * 1024) = *(const v4f*)(g + i * 1024);
#endif
    }
    __syncthreads();

    // ---- Fold: y[n] = x[n] + x[512-n]  (8 threads per row, 32 rows) -------
    {
        const int r  = tid >> 3;           // row 0..31
        const int s  = tid & 7;            // 32-col segment
        const float* xr = sX + r * NFFT;
        float*       yr = sY + r * KFOLD;
        const int n0 = s * 32;
#pragma unroll
        for (int i = 0; i < 32; ++i) {
            int n = n0 + i;
            yr[n] = xr[n] + (n ? xr[NFFT - n] : 0.0f);
        }
        if (tid < BLK_M) {                 // rank-1 row + zero pad, row = tid
            float* yt = sY + tid * KFOLD;
            yt[256] = sX[tid * NFFT + 256];
            yt[257] = 0.0f; yt[258] = 0.0f; yt[259] = 0.0f;
        }

        // ---- Column 256: out[m][256] = sum_n (-1)^n x[m][n] ---------------
        float p = 0.0f;
        const float* xa = xr + s * 64;
#pragma unroll
        for (int i = 0; i < 64; i += 2) p += xa[i] - xa[i + 1];
#pragma unroll
        for (int mask = 4; mask; mask >>= 1) p += __shfl_xor(p, mask, 8);
        if (s == 0) out[(size_t)(row_base + r) * NFFT + 256] = p;
    }
    __syncthreads();

    // ---- WMMA GEMM: 2 M-tiles x 2 N-tiles per wave over folded K=260 ------
    const int wave   = tid >> 5;       // 0..7
    const int lane   = tid & 31;
    const int h      = lane >> 4;      // half-wave (K+2 for A, row pair for B/C)
    const int m      = lane & 15;
    const int n_wave = wave * 32;

    const float* aptr0 = sY + m        * KFOLD + 2 * h;   // M rows 0..15
    const float* aptr1 = sY + (16 + m) * KFOLD + 2 * h;   // M rows 16..31
    const int c0 = (n_wave +  0 + m) * 2;                 // contiguous B frags
    const int c1 = (n_wave + 16 + m) * 2;

    v8f acc00 = {0.f,0.f,0.f,0.f,0.f,0.f,0.f,0.f};
    v8f acc01 = {0.f,0.f,0.f,0.f,0.f,0.f,0.f,0.f};
    v8f acc10 = {0.f,0.f,0.f,0.f,0.f,0.f,0.f,0.f};
    v8f acc11 = {0.f,0.f,0.f,0.f,0.f,0.f,0.f,0.f};

#pragma unroll 5
    for (int k0 = 0; k0 < KFOLD; k0 += 4) {
        v2f a0 = *(const v2f*)(aptr0 + k0);                            // LDS
        v2f a1 = *(const v2f*)(aptr1 + k0);                            // LDS
        const float* bb = Bp + ((((k0 >> 2) << 1) + h) << 9);
        v2f b0 = *(const v2f*)(bb + c0);                               // L2
        v2f b1 = *(const v2f*)(bb + c1);                               // L2
        acc00 = __builtin_amdgcn_wmma_f32_16x16x4_f32(false, a0, false, b0, (short)0, acc00, false, false);
        acc01 = __builtin_amdgcn_wmma_f32_16x16x4_f32(false, a0, false, b1, (short)0, acc01, false, false);
        acc10 = __builtin_amdgcn_wmma_f32_16x16x4_f32(false, a1, false, b0, (short)0, acc10, false, false);
        acc11 = __builtin_amdgcn_wmma_f32_16x16x4_f32(false, a1, false, b1, (short)0, acc11, false, false);
    }

    // ---- Stores: direct + mirrored cols, branchless -----------------------
    // C/D layout: VGPR v, lanes 0-15 -> M=v, N=lane; lanes 16-31 -> M=v+8.
    // Mirror (512-c)&511: lane c==0 harmlessly re-stores col 0 (same value).
    const int colA  = n_wave + m;                  // 0..255
    const int colB  = n_wave + 16 + m;             // 16..255, never 0
    const int mColA = (NFFT - colA) & (NFFT - 1);  // 257..511 (or 0 dup)
    const int mColB = NFFT - colB;                 // 257..496
    float* o0 = out + (size_t)(row_base      + 8 * h) * NFFT;   // M-tile 0
    float* o1 = out + (size_t)(row_base + 16 + 8 * h) * NFFT;   // M-tile 1
#pragma unroll
    for (int v = 0; v < 8; ++v) {
        float* p0 = o0 + (size_t)v * NFFT;
        float* p1 = o1 + (size_t)v * NFFT;
        p0[colA]  = acc00[v];  p0[mColA] = acc00[v];
        p0[colB]  = acc01[v];  p0[mColB] = acc01[v];
        p1[colA]  = acc10[v];  p1[mColA] = acc10[v];
        p1[colB]  = acc11[v];  p1[mColB] = acc11[v];
    }
}

extern "C" void kernel_launch(void* const* d_in, const int* in_sizes, int n_in,
                              void* d_out, int out_size, void* d_ws, size_t ws_size,
                              hipStream_t stream) {
    const float* A   = (const float*)d_in[0];
    float*       out = (float*)d_out;
    float*       Bp  = (float*)d_ws;   // needs 66560*4 = 266 KB of workspace

    // 66560 coefficient elements, one per thread (260 blocks * 256 = exact)
    build_dft_b_63977832841442<<<260, 256, 0, stream>>>(Bp);

    // 32768 rows / 32 rows-per-block = 1024 blocks
    dft_wmma_63977832841442<<<ROWS_TOTAL / BLK_M, 256, 0, stream>>>(A, Bp, out);
}